// PositionEnhancedLoss_58841051955399
// MI455X (gfx1250) — compile-verified
//
#include <hip/hip_runtime.h>
#include <hip/hip_bf16.h>

// PositionEnhancedLoss for MI455X (gfx1250, wave32).
// Pure streaming L1 reduction: 2 x 201 MB fp32 reads @ 23.3 TB/s => ~17us floor.
// Design:
//  - each wave owns one 512-float image row: row index / batch index / crop
//    window are wave-uniform -> scalarized (readfirstlane => SALU + s_load),
//    row mask is a uniform scalar branch (zero per-lane cost)
//  - lanes stream the row with b128 non-temporal loads (4 vec4 per lane/row)
//  - column mask = single unsigned range compare per element
//  - wave reduction via V_WMMA_F32_16X16X4_F32 (exact fp32 MAC)
//  - deterministic two-pass reduction through d_ws (no float atomics)

typedef __attribute__((ext_vector_type(4))) float vf4;
typedef __attribute__((ext_vector_type(2))) float v2f;
typedef __attribute__((ext_vector_type(8))) float v8f;

#define PEL_BLOCKS 2048
#define PEL_TPB    256
#define PEL_W      512          // image width/height
#define PEL_LEN    96           // patch length
#define PEL_HALF   48

// ---- crop_dim: window [top, bot) of width 96 clamped into [0, 512) ----
__device__ __forceinline__ void pel_crop(int d, int& top, int& bot) {
    int t = d - PEL_HALF;
    int top_rest = max(-t, 0);
    t = max(t, 0);
    int b = d + PEL_HALF;
    int bot_rest = max(b - (PEL_W - 1), 0);
    b = min(b, PEL_W - 1);
    top = t - bot_rest;
    bot = b + top_rest;
}

// ---- wave32 sum via V_WMMA_F32_16X16X4_F32 ----
// A: VGPR0 = v, VGPR1 = 0  => A[m][0]=v(m), A[m][2]=v(m+16), A[m][1]=A[m][3]=0
// B: all ones (4x16)       => D[m][n] = v(m) + v(m+16) for every column n
// d[r] holds row r (lanes 0-15) / row r+8 (lanes 16-31); sum_r d[r] gives the
// two half-wave sums, one xor-16 shuffle completes the 32-lane sum (exact).
__device__ __forceinline__ float pel_wave_sum(float v) {
    v2f a; a[0] = v;    a[1] = 0.0f;
    v2f b; b[0] = 1.0f; b[1] = 1.0f;
    v8f c = {};
    v8f d = __builtin_amdgcn_wmma_f32_16x16x4_f32(
        /*neg_a=*/false, a, /*neg_b=*/false, b,
        /*c_mod=*/(short)0, c, /*reuse_a=*/false, /*reuse_b=*/false);
    float s = ((d[0] + d[1]) + (d[2] + d[3])) + ((d[4] + d[5]) + (d[6] + d[7]));
    s += __shfl_xor(s, 16, 32);
    return s;
}

__global__ void __launch_bounds__(PEL_TPB)
pel_partial_kernel(const vf4* __restrict__ x4, const vf4* __restrict__ y4,
                   const int* __restrict__ pos,
                   float* __restrict__ part_tot, float* __restrict__ part_pat,
                   int nrows) {
    const int lane   = threadIdx.x & 31;
    const int gwave  = (blockIdx.x * PEL_TPB + threadIdx.x) >> 5;
    const int nwaves = (gridDim.x * PEL_TPB) >> 5;
    const int wl4    = lane << 2;           // this lane's float column base in chunk 0

    float tot = 0.0f, pat = 0.0f;

    for (int row = gwave; row < nrows; row += nwaves) {
        // ---- wave-uniform scalar work (SALU + s_load) ----
        const int urow = __builtin_amdgcn_readfirstlane(row);
        const int h    = urow & (PEL_W - 1);
        const int bb   = (urow >> 9) / 3;        // batch index (row / (512*3))
        const int py   = pos[2 * bb];            // rows window  (dim 2)
        const int px   = pos[2 * bb + 1];        // cols window  (dim 3)
        int ty, by, tx, bx;
        pel_crop(py, ty, by);
        pel_crop(px, tx, bx);
        const bool row_in = (h >= ty) & (h < by);  // scalar branch condition

        const vf4* __restrict__ xr = x4 + (size_t)urow * (PEL_W / 4);
        const vf4* __restrict__ yr = y4 + (size_t)urow * (PEL_W / 4);

        #pragma unroll
        for (int j = 0; j < 4; ++j) {            // 4 chunks of 32 vec4 cover the row
            const int vi = j * 32 + lane;
            const vf4 xv = __builtin_nontemporal_load(&xr[vi]);
            const vf4 yv = __builtin_nontemporal_load(&yr[vi]);
            const float l0 = fabsf(xv[0] - yv[0]);
            const float l1 = fabsf(xv[1] - yv[1]);
            const float l2 = fabsf(xv[2] - yv[2]);
            const float l3 = fabsf(xv[3] - yv[3]);
            tot += (l0 + l1) + (l2 + l3);

            if (row_in) {                        // uniform: no lane divergence
                const int d0 = (j * 128 + wl4) - tx;   // window is [tx, tx+96)
                float p = 0.0f;
                p += ((unsigned)(d0    ) < (unsigned)PEL_LEN) ? l0 : 0.0f;
                p += ((unsigned)(d0 + 1) < (unsigned)PEL_LEN) ? l1 : 0.0f;
                p += ((unsigned)(d0 + 2) < (unsigned)PEL_LEN) ? l2 : 0.0f;
                p += ((unsigned)(d0 + 3) < (unsigned)PEL_LEN) ? l3 : 0.0f;
                pat += p;
            }
        }
    }

    // All threads reconverged: EXEC all-1s for WMMA.
    const float wt = pel_wave_sum(tot);
    const float wp = pel_wave_sum(pat);

    __shared__ float st[PEL_TPB / 32];
    __shared__ float sp[PEL_TPB / 32];
    const int wid = threadIdx.x >> 5;
    if (lane == 0) { st[wid] = wt; sp[wid] = wp; }
    __syncthreads();
    if (threadIdx.x == 0) {
        float a = 0.0f, b = 0.0f;
        #pragma unroll
        for (int i = 0; i < PEL_TPB / 32; ++i) { a += st[i]; b += sp[i]; }
        part_tot[blockIdx.x] = a;
        part_pat[blockIdx.x] = b;
    }
}

__global__ void __launch_bounds__(PEL_TPB)
pel_final_kernel(const float* __restrict__ part_tot,
                 const float* __restrict__ part_pat,
                 float* __restrict__ out) {
    float t = 0.0f, p = 0.0f;
    for (int i = threadIdx.x; i < PEL_BLOCKS; i += PEL_TPB) {
        t += part_tot[i];
        p += part_pat[i];
    }
    const float wt = pel_wave_sum(t);
    const float wp = pel_wave_sum(p);

    __shared__ float st[PEL_TPB / 32];
    __shared__ float sp[PEL_TPB / 32];
    const int lane = threadIdx.x & 31;
    const int wid  = threadIdx.x >> 5;
    if (lane == 0) { st[wid] = wt; sp[wid] = wp; }
    __syncthreads();
    if (threadIdx.x == 0) {
        float T = 0.0f, P = 0.0f;
        #pragma unroll
        for (int i = 0; i < PEL_TPB / 32; ++i) { T += st[i]; P += sp[i]; }
        // patch_numel = 64*3*96*96 = 1,769,472 ; total numel = 50,331,648
        const float patch_numel = 1769472.0f;
        const float rest_numel  = 50331648.0f - 1769472.0f;   // 48,562,176
        const float patch_loss  = P / patch_numel;
        const float rest_loss   = (T - P) / rest_numel;
        out[0] = 0.5f * rest_loss + 0.5f * patch_loss;        // FACTOR = 0.5
    }
}

extern "C" void kernel_launch(void* const* d_in, const int* in_sizes, int n_in,
                              void* d_out, int out_size, void* d_ws, size_t ws_size,
                              hipStream_t stream) {
    const float* x = (const float*)d_in[0];
    const float* y = (const float*)d_in[1];
    const int* pos = (const int*)d_in[2];
    float* out     = (float*)d_out;

    const int n     = in_sizes[0];     // 50,331,648
    const int nrows = n >> 9;          // rows of 512 floats: 98,304

    float* part_tot = (float*)d_ws;
    float* part_pat = part_tot + PEL_BLOCKS;

    pel_partial_kernel<<<PEL_BLOCKS, PEL_TPB, 0, stream>>>(
        (const vf4*)x, (const vf4*)y, pos, part_tot, part_pat, nrows);
    pel_final_kernel<<<1, PEL_TPB, 0, stream>>>(part_tot, part_pat, out);
}